// ReblurLayer_58737972740753
// MI455X (gfx1250) — compile-verified
//
#include <hip/hip_runtime.h>
#include <cstdint>
#include <cstddef>

// ---------------------------------------------------------------------------
// Per-pixel dynamic convolution (K=13) for MI455X / gfx1250.
//
// Bandwidth-bound streaming kernel (~16 us roofline at 23.3 TB/s):
//  - 354 MB per-pixel weight tensor streamed ONCE with non-temporal
//    global_load_b128 (th:TH_LOAD_NT keeps L2 free for the reused image).
//  - Image tile staged into LDS via CDNA5 async global->LDS copies
//    (global_load_async_to_lds_b32, ASYNCcnt, s_wait_asynccnt).
//  - 16-float sliding window register-cached per ki; inner loop is pure
//    v_fmac_f32 on VGPRs (12 FMAs per 16B weight load).
//  - Output written once with non-temporal stores.
// ---------------------------------------------------------------------------

namespace {
constexpr int KS     = 13;
constexpr int PADR   = KS / 2;          // 6
constexpr int Bn     = 8;
constexpr int Cn     = 3;
constexpr int Hn     = 256;
constexpr int Wn     = 256;
constexpr int TILE_H = 16;
constexpr int TILE_W = 64;
constexpr int IN_H   = TILE_H + KS - 1; // 28
constexpr int IN_W   = TILE_W + KS - 1; // 76
constexpr int IN_WP  = 80;              // padded row: 320 B, keeps 16B alignment
constexpr int SMEM_N = Cn * IN_H * IN_WP; // 6720 floats = 26.9 KB
constexpr int NTHREADS = 256;
}

// Native clang vector type: required by __builtin_nontemporal_load/store
// (HIP's float4 is a C++ class, which the builtin rejects).
typedef __attribute__((ext_vector_type(4))) float f32x4;

typedef __attribute__((address_space(3))) float lds_f32;

// LDS byte offset of a __shared__ element (generic -> AS(3) -> int strips the
// aperture; async-to-LDS VDST wants the raw LDS byte address).
__device__ __forceinline__ uint32_t lds_byte_off(const float* p) {
  return (uint32_t)(uintptr_t)(lds_f32*)p;
}

// CDNA5 async global->LDS copy, one dword per lane. Tracked by ASYNCcnt.
__device__ __forceinline__ void async_g2lds_b32(uint32_t ldsoff, const float* g) {
  asm volatile("global_load_async_to_lds_b32 %0, %1, off"
               :
               : "v"(ldsoff), "v"((unsigned long long)(uintptr_t)g)
               : "memory");
}

__device__ __forceinline__ void wait_asynccnt0() {
  asm volatile("s_wait_asynccnt 0" ::: "memory");
}

__global__ __launch_bounds__(NTHREADS)
void reblur_dynconv_kernel(const float* __restrict__ image,
                           const float* __restrict__ kern,
                           float* __restrict__ out) {
  __shared__ __align__(16) float smem[SMEM_N];

  const int tid = threadIdx.x;
  const int b   = blockIdx.z;
  const int h0  = blockIdx.y * TILE_H;
  const int w0  = blockIdx.x * TILE_W;

  const float* img_b = image + (size_t)b * Cn * Hn * Wn;

  // ---- Stage padded image tile (3 x 28 x 80) into LDS.
  // In-range elements: async DMA global->LDS. Out-of-range (halo): zero.
  for (int i = tid; i < SMEM_N; i += NTHREADS) {
    const int c  = i / (IN_H * IN_WP);
    const int r  = (i / IN_WP) % IN_H;
    const int cc = i % IN_WP;
    const int gr = h0 - PADR + r;
    const int gc = w0 - PADR + cc;
    if (cc < IN_W && gr >= 0 && gr < Hn && gc >= 0 && gc < Wn) {
      async_g2lds_b32(lds_byte_off(&smem[i]),
                      img_b + ((size_t)c * Hn + gr) * Wn + gc);
    } else {
      smem[i] = 0.0f;   // ds_store_b32 zero for halo/padding
    }
  }
  wait_asynccnt0();
  __syncthreads();

  // ---- Each thread owns 4 consecutive output pixels along w (float4 streams).
  const int tx4 = (tid & 15) * 4;   // 0..60
  const int ty  = tid >> 4;         // 0..15
  const int hO  = h0 + ty;
  const int wO  = w0 + tx4;
  const size_t pix = (size_t)hO * Wn + wO;

  // kernel layout: [b][0][ki][kj][h][w]; tap plane stride = H*W
  const float* kb = kern + (size_t)b * (KS * KS) * Hn * Wn + pix;

  float acc[Cn][4];
#pragma unroll
  for (int c = 0; c < Cn; ++c)
#pragma unroll
    for (int p = 0; p < 4; ++p) acc[c][p] = 0.0f;

  for (int ki = 0; ki < KS; ++ki) {
    // 16-float sliding window per channel (covers kj=0..12 for 4 pixels),
    // loaded as 4 aligned ds_load_b128 per channel, then held in VGPRs.
    float win[Cn][16];
#pragma unroll
    for (int c = 0; c < Cn; ++c) {
      const f32x4* p4 = reinterpret_cast<const f32x4*>(
          &smem[((c * IN_H) + ty + ki) * IN_WP + tx4]);
#pragma unroll
      for (int v = 0; v < 4; ++v) {
        const f32x4 t = p4[v];
        win[c][4 * v + 0] = t.x;
        win[c][4 * v + 1] = t.y;
        win[c][4 * v + 2] = t.z;
        win[c][4 * v + 3] = t.w;
      }
    }

#pragma unroll
    for (int kj = 0; kj < KS; ++kj) {
      // One coalesced, NON-TEMPORAL 16B weight load feeds 12 FMAs.
      // Weights are touched exactly once -> th:TH_LOAD_NT so they don't
      // evict the reused image tiles from L2.
      const f32x4 kv = __builtin_nontemporal_load(
          reinterpret_cast<const f32x4*>(
              kb + (size_t)(ki * KS + kj) * (Hn * Wn)));
#pragma unroll
      for (int c = 0; c < Cn; ++c) {
        acc[c][0] = fmaf(kv.x, win[c][kj + 0], acc[c][0]);
        acc[c][1] = fmaf(kv.y, win[c][kj + 1], acc[c][1]);
        acc[c][2] = fmaf(kv.z, win[c][kj + 2], acc[c][2]);
        acc[c][3] = fmaf(kv.w, win[c][kj + 3], acc[c][3]);
      }
    }
  }

  // ---- Write 3 channels x float4, non-temporal (never re-read).
#pragma unroll
  for (int c = 0; c < Cn; ++c) {
    f32x4 o;
    o.x = acc[c][0]; o.y = acc[c][1]; o.z = acc[c][2]; o.w = acc[c][3];
    __builtin_nontemporal_store(
        o, reinterpret_cast<f32x4*>(
               out + ((size_t)b * Cn + c) * Hn * Wn + pix));
  }
}

extern "C" void kernel_launch(void* const* d_in, const int* in_sizes, int n_in,
                              void* d_out, int out_size, void* d_ws, size_t ws_size,
                              hipStream_t stream) {
  (void)in_sizes; (void)n_in; (void)out_size; (void)d_ws; (void)ws_size;
  const float* image = (const float*)d_in[0];  // (8,3,256,256) f32
  const float* kern  = (const float*)d_in[1];  // (8,1,13,13,256,256) f32
  float* out = (float*)d_out;                  // (8,3,256,256) f32

  dim3 grid(Wn / TILE_W, Hn / TILE_H, Bn);     // (4,16,8) = 512 workgroups
  dim3 block(NTHREADS);
  reblur_dynconv_kernel<<<grid, block, 0, stream>>>(image, kern, out);
}